// SSIMLoss_14018773254236
// MI455X (gfx1250) — compile-verified
//
#include <hip/hip_runtime.h>

// ---------------------------------------------------------------------------
// Fused SSIM loss for 16x3x512x512 fp32, MI455X (gfx1250, wave32).
// Separable 11-tap Gaussian depthwise conv done with V_WMMA_F32_16X16X4_F32
// as banded-matrix products; fully fused (zero intermediate global tensors).
// ---------------------------------------------------------------------------

typedef float v2f __attribute__((ext_vector_type(2)));
typedef float v8f __attribute__((ext_vector_type(8)));

#define TILE   16
#define HALO   5
#define EXT    26          // TILE + 2*HALO
#define PSTR   27          // LDS stride for 26-wide input tiles (odd -> bank spread)
#define HSTR   17          // LDS stride for h-blur intermediate
#define PW_SIZE (2*EXT*PSTR + EXT*HSTR + 6)   // per-wave LDS floats (1852)
#define WPB    8           // waves (tiles) per 256-thread block
#define IMG    512
#define PLANES (16*3)
#define TILES_PER_PLANE (32*32)
#define NTILES (PLANES*TILES_PER_PLANE)       // 49152
#define NBLOCKS (NTILES/WPB)                  // 6144

__device__ __forceinline__ float gauss_unnorm(int idx) {
    float d = (float)(idx - 5);
    return __expf(-d * d * (1.0f / 4.5f));    // sigma = 1.5 -> 2*sigma^2 = 4.5
}

__global__ __launch_bounds__(256) void ssim_tile_kernel(
    const float* __restrict__ pred,
    const float* __restrict__ targ,
    float* __restrict__ partial)
{
    __shared__ float smem[WPB * PW_SIZE];
    __shared__ float wavesum[WPB];

    const int lane = threadIdx.x & 31;
    const int wave = threadIdx.x >> 5;
    float* sp = smem + wave * PW_SIZE;              // pred halo tile  [26][27]
    float* st = sp + EXT * PSTR;                    // target halo tile[26][27]
    float* sh = st + EXT * PSTR;                    // h-blur scratch  [26][17]

    // ---- tile assignment: one 16x16 output tile per wave ----
    const int tile  = blockIdx.x * WPB + wave;      // < 49152
    const int plane = tile >> 10;
    const int trem  = tile & 1023;
    const int ty    = trem >> 5;
    const int tx    = trem & 31;
    const long base = (long)plane * (IMG * IMG);
    const int r0 = ty * TILE - HALO;
    const int c0 = tx * TILE - HALO;

    // ---- cooperative load of 26x26 halos (zero-padded, matches SAME conv) ----
    for (int e = lane; e < EXT * EXT; e += 32) {
        int rr = e / EXT, cc = e - rr * EXT;
        int gr = r0 + rr, gc = c0 + cc;
        bool ok = (gr >= 0) && (gr < IMG) && (gc >= 0) && (gc < IMG);
        long idx = base + (long)gr * IMG + gc;
        sp[rr * PSTR + cc] = ok ? pred[idx] : 0.0f;
        st[rr * PSTR + cc] = ok ? targ[idx] : 0.0f;
    }
    // wave-private LDS + in-order DS per wave: no cross-wave barrier needed.

    // ---- normalized Gaussian weight helper state ----
    float wsum = 0.0f;
#pragma unroll
    for (int i = 0; i < 11; ++i) wsum += gauss_unnorm(i);
    const float invS = 1.0f / wsum;

    const int half = lane >> 4;       // ISA fragment half select
    const int nidx = lane & 15;       // N (B/C/D) or M (A) index
    const int koff = half * 2;        // k = j + 2*(lane>=16)

    // ---- H-pass weight fragments: B[k][n] = g[(4c+k) - n]  (banded) ----
    v2f hB[7];
#pragma unroll
    for (int c = 0; c < 7; ++c) {
#pragma unroll
        for (int j = 0; j < 2; ++j) {
            int kcol = 4 * c + koff + j;
            int d = kcol - nidx;
            hB[c][j] = (d >= 0 && d <= 10 && kcol < EXT)
                         ? gauss_unnorm(d) * invS : 0.0f;
        }
    }
    // ---- V-pass weight fragments: A[m][k] = g[(4c+k) - m] ----
    v2f vA[7];
#pragma unroll
    for (int c = 0; c < 7; ++c) {
#pragma unroll
        for (int j = 0; j < 2; ++j) {
            int krow = 4 * c + koff + j;
            int d = krow - nidx;
            vA[c][j] = (d >= 0 && d <= 10 && krow < EXT)
                         ? gauss_unnorm(d) * invS : 0.0f;
        }
    }

    // ---- horizontal pass: 5 fields x 2 row-groups, accumulate over 7 chunks ----
    v8f hd[5][2];
#pragma unroll
    for (int f = 0; f < 5; ++f) { v8f z = {}; hd[f][0] = z; hd[f][1] = z; }

#pragma unroll
    for (int rg = 0; rg < 2; ++rg) {
        const int R = rg * 10;                       // rows 0-15 and 10-25
#pragma unroll
        for (int c = 0; c < 7; ++c) {
            v2f aP, aT, aPP, aTT, aPT;
#pragma unroll
            for (int j = 0; j < 2; ++j) {
                int col = 4 * c + koff + j;
                int row = R + nidx;
                float p = (col < EXT) ? sp[row * PSTR + col] : 0.0f;
                float t = (col < EXT) ? st[row * PSTR + col] : 0.0f;
                aP[j] = p; aT[j] = t; aPP[j] = p * p; aTT[j] = t * t; aPT[j] = p * t;
            }
            hd[0][rg] = __builtin_amdgcn_wmma_f32_16x16x4_f32(false, aP,  false, hB[c], (short)0, hd[0][rg], false, false);
            hd[1][rg] = __builtin_amdgcn_wmma_f32_16x16x4_f32(false, aT,  false, hB[c], (short)0, hd[1][rg], false, false);
            hd[2][rg] = __builtin_amdgcn_wmma_f32_16x16x4_f32(false, aPP, false, hB[c], (short)0, hd[2][rg], false, false);
            hd[3][rg] = __builtin_amdgcn_wmma_f32_16x16x4_f32(false, aTT, false, hB[c], (short)0, hd[3][rg], false, false);
            hd[4][rg] = __builtin_amdgcn_wmma_f32_16x16x4_f32(false, aPT, false, hB[c], (short)0, hd[4][rg], false, false);
        }
    }

    // ---- vertical pass per field (round-trip h-blur through LDS) ----
    v8f acc[5];
#pragma unroll
    for (int f = 0; f < 5; ++f) {
        // D layout: vgpr i -> row i (lanes 0-15) / row i+8 (lanes 16-31), col = nidx
#pragma unroll
        for (int i = 0; i < 8; ++i) {
            int row = i + half * 8;
            sh[row * HSTR + nidx] = hd[f][0][i];          // rows 0..15
        }
#pragma unroll
        for (int i = 0; i < 8; ++i) {
            int row = 10 + i + half * 8;
            sh[row * HSTR + nidx] = hd[f][1][i];          // rows 10..25 (overlap identical)
        }
        v8f z = {}; acc[f] = z;
#pragma unroll
        for (int c = 0; c < 7; ++c) {
            v2f b;
#pragma unroll
            for (int j = 0; j < 2; ++j) {
                int row = 4 * c + koff + j;
                b[j] = (row < EXT) ? sh[row * HSTR + nidx] : 0.0f;
            }
            acc[f] = __builtin_amdgcn_wmma_f32_16x16x4_f32(false, vA[c], false, b, (short)0, acc[f], false, false);
        }
    }

    // ---- elementwise SSIM on the 8 D-fragment elements + reduction ----
    const float C1 = 0.0001f;   // 0.01^2
    const float C2 = 0.0009f;   // 0.03^2
    float lsum = 0.0f;
#pragma unroll
    for (int i = 0; i < 8; ++i) {
        float mu1 = acc[0][i], mu2 = acc[1][i];
        float mu1s = mu1 * mu1, mu2s = mu2 * mu2, mu12 = mu1 * mu2;
        float s11 = acc[2][i] - mu1s;
        float s22 = acc[3][i] - mu2s;
        float s12 = acc[4][i] - mu12;
        float num = (2.0f * mu12 + C1) * (2.0f * s12 + C2);
        float den = (mu1s + mu2s + C1) * (s11 + s22 + C2);
        lsum += num / den;
    }
    // wave32 butterfly reduce
#pragma unroll
    for (int off = 16; off > 0; off >>= 1)
        lsum += __shfl_xor(lsum, off, 32);
    if (lane == 0) wavesum[wave] = lsum;
    __syncthreads();
    if (threadIdx.x == 0) {
        float s = 0.0f;
#pragma unroll
        for (int w = 0; w < WPB; ++w) s += wavesum[w];   // deterministic order
        partial[blockIdx.x] = s;
    }
}

__global__ __launch_bounds__(256) void ssim_reduce_kernel(
    const float* __restrict__ partial, int n, float* __restrict__ out)
{
    __shared__ float sm[256];
    float s = 0.0f;
    for (int i = threadIdx.x; i < n; i += 256) s += partial[i];
    sm[threadIdx.x] = s;
    __syncthreads();
    for (int stride = 128; stride > 0; stride >>= 1) {
        if (threadIdx.x < stride) sm[threadIdx.x] += sm[threadIdx.x + stride];
        __syncthreads();
    }
    if (threadIdx.x == 0) {
        const float N = (float)PLANES * (float)(IMG) * (float)(IMG);
        out[0] = 1.0f - sm[0] / N;                       // loss_weight = 1.0
    }
}

extern "C" void kernel_launch(void* const* d_in, const int* in_sizes, int n_in,
                              void* d_out, int out_size, void* d_ws, size_t ws_size,
                              hipStream_t stream) {
    (void)in_sizes; (void)n_in; (void)out_size; (void)ws_size;
    const float* pred = (const float*)d_in[0];
    const float* targ = (const float*)d_in[1];
    float* partial = (float*)d_ws;                       // NBLOCKS floats of scratch
    ssim_tile_kernel<<<NBLOCKS, 256, 0, stream>>>(pred, targ, partial);
    ssim_reduce_kernel<<<1, 256, 0, stream>>>(partial, NBLOCKS, (float*)d_out);
}